// HOTFormerStage_69681549410363
// MI455X (gfx1250) — compile-verified
//
#include <hip/hip_runtime.h>

// ---------------------------------------------------------------------------
// HOTFormer HAT window-attention stage, fused per-window kernel for gfx1250.
// N=4096 windows, T=33 tokens (32 + 1 relay), C=96, H=6 heads, HD=16.
// GEMMs on V_WMMA_F32_16X16X4_F32 (fp32 in / fp32 acc, matches reference).
// Data/mask staged into LDS by the Tensor Data Mover (iterate mode), RPE
// gather runs concurrently on all waves.
// ---------------------------------------------------------------------------

#define NWIN   4096
#define KTOK   32
#define TTOK   33            // K + G
#define CCH    96
#define HEADS  6
#define RPE_N  51
#define QSCALE 0.25f         // HD^-0.5, HD=16

#define TP        48         // tokens padded to 3x16
#define SD_STRIDE 100        // sData / sOut row stride (floats, even)
#define SQ_STRIDE 292        // sQKV row stride (288 cols + pad, even)
#define SS_STRIDE 38         // per-head prob row stride (36 cols + pad, even)
#define SM_STRIDE 36         // mask row stride

typedef __attribute__((ext_vector_type(2))) float        v2f;
typedef __attribute__((ext_vector_type(8))) float        v8f;
typedef __attribute__((ext_vector_type(4))) unsigned int u32x4;
typedef __attribute__((ext_vector_type(8))) int          i32x8;
typedef __attribute__((ext_vector_type(4))) int          i32x4;

#if defined(__has_builtin)
#if __has_builtin(__builtin_amdgcn_tensor_load_to_lds) && \
    __has_builtin(__builtin_amdgcn_s_wait_tensorcnt)
#define USE_TDM 1
#endif
#endif
#ifndef USE_TDM
#define USE_TDM 0
#endif

__device__ __forceinline__ v8f wmma_k4(v2f a, v2f b, v8f c) {
  // D = A(16x4) * B(4x16) + C(16x16), fp32
  return __builtin_amdgcn_wmma_f32_16x16x4_f32(false, a, false, b,
                                               (short)0, c, false, false);
}

#if USE_TDM
// TDM 2D strided-row copy via iterate mode: copies `rows` rows of
// `row_elems` fp32 each; LDS rows padded to lds_stride elems.
// D# packing per CDNA5 ISA 8.3-8.5 (group0/1/2; group3 unused -> zeros).
__device__ __forceinline__ void tdm_load_rows(void* lds_ptr, const void* gptr,
                                              int row_elems, int rows,
                                              int lds_stride, int g_stride) {
  unsigned long long ga = (unsigned long long)gptr;
  unsigned int lds_off  = (unsigned int)(unsigned long long)lds_ptr; // low32 = LDS offset

  u32x4 g0;
  g0[0] = 1u;                                          // count=1, user mode
  g0[1] = lds_off;                                     // lds_addr (bytes)
  g0[2] = (unsigned int)(ga & 0xFFFFFFFFull);          // global_addr[31:0]
  g0[3] = (unsigned int)((ga >> 32) & 0x01FFFFFFull)   // global_addr[56:32]
        | (2u << 30);                                  // type = 2 ("image")

  i32x8 g1;
  g1[0] = (2 << 16) | (1 << 19);                       // data_size=4B, iterate_enable
  g1[1] = (row_elems & 0xFFFF) << 16;                  // tensor_dim0[15:0] @ [63:48]
  g1[2] = ((row_elems >> 16) & 0xFFFF)                 // tensor_dim0[31:16]
        | ((rows & 0xFFFF) << 16);                     // tensor_dim1[15:0]
  g1[3] = ((rows >> 16) & 0xFFFF)                      // tensor_dim1[31:16]
        | ((row_elems & 0xFFFF) << 16);                // tile_dim0
  g1[4] = 1;                                           // tile_dim1=1, tile_dim2=0
  g1[5] = g_stride;                                    // tensor_dim0_stride[31:0]
  g1[6] = 0;                                           // stride hi | dim1_stride lo
  g1[7] = 0;

  i32x4 g2;
  g2[0] = 0;                                           // tensor_dim2
  g2[1] = lds_stride;                                  // lds_addr_increment (elems)
  g2[2] = g_stride;                                    // global_addr_increment (elems)
  g2[3] = ((rows - 1) & 0xFF) << 16;                   // iterate_count @ [127:112]

  i32x4 g3;
  g3[0] = 0; g3[1] = 0; g3[2] = 0; g3[3] = 0;

#if defined(__clang_major__) && (__clang_major__ >= 23)
  i32x8 gz;
  gz[0]=0; gz[1]=0; gz[2]=0; gz[3]=0; gz[4]=0; gz[5]=0; gz[6]=0; gz[7]=0;
  __builtin_amdgcn_tensor_load_to_lds(g0, g1, g2, g3, gz, 0);
#else
  __builtin_amdgcn_tensor_load_to_lds(g0, g1, g2, g3, 0);
#endif
}
#endif // USE_TDM

__global__ __launch_bounds__(256, 2)
void hat_attn_kernel(const float* __restrict__ data,      // [N,33,96]
                     const int*   __restrict__ rel_pos,   // [N,32,32,3]
                     const float* __restrict__ mask,      // [N,33,33]
                     const float* __restrict__ qkv_w,     // [96,288]
                     const float* __restrict__ qkv_b,     // [288]
                     const float* __restrict__ proj_w,    // [96,96]
                     const float* __restrict__ proj_b,    // [96]
                     const float* __restrict__ rpe_table, // [153,6]
                     float* __restrict__ out)             // [N,33,96]
{
  __shared__ float sData[TP * SD_STRIDE];            // input tile; reused as attn-out
  __shared__ float sQKV [TP * SQ_STRIDE];            // [48][q|k|v = 288]
  __shared__ float sRpe [KTOK * KTOK * HEADS];       // [32][32][6]
  __shared__ float sMask[TTOK * SM_STRIDE];          // [33][36]
  __shared__ float sS   [HEADS * TP * SS_STRIDE];    // per-head probs [48][38]

  const int win  = blockIdx.x;
  const int tid  = threadIdx.x;
  const int wv   = tid >> 5;      // wave id 0..7
  const int lane = tid & 31;
  const int lo   = lane & 15;     // column / M-row selector inside fragment
  const int hi   = lane >> 4;     // K-half selector

  // ---------------- Phase A: stage data + mask (TDM), RPE gather -------------
  const float* dN = data + (size_t)win * (TTOK * CCH);
  const float* mN = mask + (size_t)win * (TTOK * TTOK);
#if USE_TDM
  if (wv == 0) {
    tdm_load_rows(sData, dN, CCH, TTOK, SD_STRIDE, CCH);     // 33 x 96 rows
    tdm_load_rows(sMask, mN, TTOK, TTOK, SM_STRIDE, TTOK);   // 33 x 33 rows
  }
  // zero the padded token rows (disjoint from TDM destination)
  for (int idx = tid; idx < (TP - TTOK) * SD_STRIDE; idx += 256)
    sData[TTOK * SD_STRIDE + idx] = 0.0f;
#else
  for (int idx = tid; idx < TP * (CCH / 4); idx += 256) {
    int row = idx / (CCH / 4);
    int c4  = (idx % (CCH / 4)) * 4;
    float4 v = make_float4(0.f, 0.f, 0.f, 0.f);
    if (row < TTOK) v = *(const float4*)(dN + row * CCH + c4);
    float* p = &sData[row * SD_STRIDE + c4];
    p[0] = v.x; p[1] = v.y; p[2] = v.z; p[3] = v.w;
  }
  for (int idx = tid; idx < TTOK * TTOK; idx += 256) {
    int r = idx / TTOK, c = idx % TTOK;
    sMask[r * SM_STRIDE + c] = mN[idx];
  }
#endif
  // rpe: clip(rel-25,-25,25)+25+axis*51 == clip(rel,0,50)+axis*51
  {
    const int* rpN = rel_pos + (size_t)win * (KTOK * KTOK * 3);
    for (int idx = tid; idx < KTOK * KTOK; idx += 256) {
      int b = idx * 3;
      int i0 = min(max(rpN[b + 0], 0), RPE_N - 1);
      int i1 = min(max(rpN[b + 1], 0), RPE_N - 1) + RPE_N;
      int i2 = min(max(rpN[b + 2], 0), RPE_N - 1) + 2 * RPE_N;
      const float* t0 = rpe_table + i0 * HEADS;
      const float* t1 = rpe_table + i1 * HEADS;
      const float* t2 = rpe_table + i2 * HEADS;
      #pragma unroll
      for (int h = 0; h < HEADS; ++h)
        sRpe[idx * HEADS + h] = t0[h] + t1[h] + t2[h];
    }
  }
#if USE_TDM
  if (wv == 0) __builtin_amdgcn_s_wait_tensorcnt(0);  // drain before barrier
#endif
  __syncthreads();

  // ---------------- Phase B: QKV GEMM [48,96] x [96,288] -> LDS --------------
  // Wave owns a 16-wide column panel; B fragment reused across 3 row tiles.
  for (int ct = wv; ct < 18; ct += 8) {
    int col0 = ct * 16;
    v8f acc[3];
    acc[0] = {}; acc[1] = {}; acc[2] = {};
    #pragma unroll
    for (int k0 = 0; k0 < CCH; k0 += 4) {
      int kk = k0 + hi * 2;
      v2f b;
      b.x = qkv_w[kk * 288 + col0 + lo];                    // B(kk ,n)
      b.y = qkv_w[(kk + 1) * 288 + col0 + lo];              // B(kk+1,n)
      #pragma unroll
      for (int i = 0; i < 3; ++i) {
        v2f a = *(const v2f*)&sData[(i * 16 + lo) * SD_STRIDE + kk];
        acc[i] = wmma_k4(a, b, acc[i]);
      }
    }
    float bias = qkv_b[col0 + lo];
    bool  isQ  = (col0 < CCH);
    #pragma unroll
    for (int i = 0; i < 3; ++i) {
      #pragma unroll
      for (int r = 0; r < 8; ++r) {
        int m = i * 16 + r + hi * 8;
        float v = acc[i][r] + bias;
        if (isQ) v *= QSCALE;
        sQKV[m * SQ_STRIDE + col0 + lo] = (m < TTOK) ? v : 0.0f;
      }
    }
  }
  __syncthreads();

  // ---------------- Phase C: per-head attention (wave h -> head h) -----------
  if (wv < HEADS) {
    const int h = wv;
    float* Sw = &sS[h * TP * SS_STRIDE];

    v8f acc[3][3];
    #pragma unroll
    for (int i = 0; i < 3; ++i)
      #pragma unroll
      for (int j = 0; j < 3; ++j) acc[i][j] = {};

    // S = (Q * scale) K^T : K-dim = HD = 16
    #pragma unroll
    for (int k0 = 0; k0 < 16; k0 += 4) {
      int kk = k0 + hi * 2;
      v2f a[3], b[3];
      #pragma unroll
      for (int i = 0; i < 3; ++i)      // Q fragment: A(m,kk) = Q[m][kk]
        a[i] = *(const v2f*)&sQKV[(i * 16 + lo) * SQ_STRIDE + h * 16 + kk];
      #pragma unroll
      for (int j = 0; j < 3; ++j)      // K^T fragment: B(kk,n) = K[n][kk]
        b[j] = *(const v2f*)&sQKV[(j * 16 + lo) * SQ_STRIDE + CCH + h * 16 + kk];
      #pragma unroll
      for (int i = 0; i < 3; ++i)
        #pragma unroll
        for (int j = 0; j < 3; ++j) acc[i][j] = wmma_k4(a[i], b[j], acc[i][j]);
    }

    // epilogue: + rpe + mask, row softmax over the 33 valid keys, spill P
    #pragma unroll
    for (int i = 0; i < 3; ++i) {
      #pragma unroll
      for (int r = 0; r < 8; ++r) {
        int m = i * 16 + r + hi * 8;
        float s[3];
        #pragma unroll
        for (int j = 0; j < 3; ++j) {
          int n = j * 16 + lo;
          float v = acc[i][j][r];
          if (n >= TTOK) {
            v = -3.0e38f;                       // pad column -> prob 0
          } else if (m < TTOK) {
            v += sMask[m * SM_STRIDE + n];
            if (m >= 1 && n >= 1)
              v += sRpe[((m - 1) * KTOK + (n - 1)) * HEADS + h];
          }
          s[j] = v;
        }
        float lm = fmaxf(fmaxf(s[0], s[1]), s[2]);
        #pragma unroll
        for (int off = 1; off < 16; off <<= 1)
          lm = fmaxf(lm, __shfl_xor(lm, off, 32));
        float e0 = __expf(s[0] - lm);
        float e1 = __expf(s[1] - lm);
        float e2 = __expf(s[2] - lm);
        float ls = e0 + e1 + e2;
        #pragma unroll
        for (int off = 1; off < 16; off <<= 1)
          ls += __shfl_xor(ls, off, 32);
        float rinv = 1.0f / ls;
        Sw[m * SS_STRIDE + lo]      = e0 * rinv;
        Sw[m * SS_STRIDE + 16 + lo] = e1 * rinv;
        if (lo < 4)                               // cols 32..35 (>=33 store 0)
          Sw[m * SS_STRIDE + 32 + lo] = e2 * rinv;
      }
    }

    // O = P V  (K truncated to 36: V rows >=33 are zero, P cols 33..35 are 0)
    #pragma unroll
    for (int i = 0; i < 3; ++i) {
      v8f o = {};
      int m = i * 16 + lo;
      #pragma unroll
      for (int k0 = 0; k0 < 36; k0 += 4) {
        int kk = k0 + hi * 2;
        v2f a = *(const v2f*)&Sw[m * SS_STRIDE + kk];        // P(m,kk..kk+1)
        v2f b;
        b.x = sQKV[kk * SQ_STRIDE + 2 * CCH + h * 16 + lo];  // V(kk ,n)
        b.y = sQKV[(kk + 1) * SQ_STRIDE + 2 * CCH + h * 16 + lo];
        o = wmma_k4(a, b, o);
      }
      #pragma unroll
      for (int r = 0; r < 8; ++r) {
        int mm = i * 16 + r + hi * 8;
        sData[mm * SD_STRIDE + h * 16 + lo] = o[r];          // sOut reuse
      }
    }
  }
  __syncthreads();

  // ---------------- Phase D: proj GEMM [48,96] x [96,96] + bias --------------
  // Wave owns a column panel; B fragment reused across 3 row tiles.
  for (int ct = wv; ct < 6; ct += 8) {
    int col0 = ct * 16;
    v8f acc[3];
    acc[0] = {}; acc[1] = {}; acc[2] = {};
    #pragma unroll
    for (int k0 = 0; k0 < CCH; k0 += 4) {
      int kk = k0 + hi * 2;
      v2f b;
      b.x = proj_w[kk * CCH + col0 + lo];
      b.y = proj_w[(kk + 1) * CCH + col0 + lo];
      #pragma unroll
      for (int i = 0; i < 3; ++i) {
        v2f a = *(const v2f*)&sData[(i * 16 + lo) * SD_STRIDE + kk];
        acc[i] = wmma_k4(a, b, acc[i]);
      }
    }
    float bias = proj_b[col0 + lo];
    #pragma unroll
    for (int i = 0; i < 3; ++i) {
      #pragma unroll
      for (int r = 0; r < 8; ++r) {
        int m = i * 16 + r + hi * 8;
        if (m < TTOK)
          out[(size_t)win * (TTOK * CCH) + m * CCH + col0 + lo] = acc[i][r] + bias;
      }
    }
  }
}

extern "C" void kernel_launch(void* const* d_in, const int* in_sizes, int n_in,
                              void* d_out, int out_size, void* d_ws, size_t ws_size,
                              hipStream_t stream) {
  (void)in_sizes; (void)n_in; (void)out_size; (void)d_ws; (void)ws_size;
  const float* data      = (const float*)d_in[0];
  const int*   rel_pos   = (const int*)  d_in[1];
  const float* mask      = (const float*)d_in[2];
  const float* qkv_w     = (const float*)d_in[3];
  const float* qkv_b     = (const float*)d_in[4];
  const float* proj_w    = (const float*)d_in[5];
  const float* proj_b    = (const float*)d_in[6];
  const float* rpe_table = (const float*)d_in[7];
  float* out = (float*)d_out;

  hipLaunchKernelGGL(hat_attn_kernel, dim3(NWIN), dim3(256), 0, stream,
                     data, rel_pos, mask, qkv_w, qkv_b, proj_w, proj_b,
                     rpe_table, out);
}